// WordPooler_56324201120274
// MI455X (gfx1250) — compile-verified
//
#include <hip/hip_runtime.h>
#include <stdint.h>

#define PAD_VAL (-100)

// Native 128-bit vector type (HIP's float4 wrapper class is rejected by
// __builtin_nontemporal_store; ext_vector_type is a true vector type).
typedef float v4f __attribute__((ext_vector_type(4)));

// ---------------------------------------------------------------------------
// Kernel 1: map[b*T + w] = -1 for all output rows (re-init every call).
// ---------------------------------------------------------------------------
__global__ void wp_init_map(int* __restrict__ map, int n) {
    int i = blockIdx.x * blockDim.x + threadIdx.x;
    if (i < n) map[i] = -1;
}

// ---------------------------------------------------------------------------
// Kernel 2: one thread per token. A token is the "first" subword of its word
// iff it is valid (!= PAD) and differs from its predecessor (offsets are
// sorted, PAD region is trailing, so != PAD <=> cumprod-valid). Each word gets
// at most one first token, so a plain store (no atomics) is race-free.
// ---------------------------------------------------------------------------
__global__ void wp_mark_first(const int* __restrict__ ofs,
                              int* __restrict__ map,
                              int S, int T, int total) {
    int t = blockIdx.x * blockDim.x + threadIdx.x;
    if (t >= total) return;
    int w = ofs[t];
    if (w == PAD_VAL) return;
    int s = t % S;
    int b = t / S;
    bool first = (s == 0) || (ofs[t - 1] != w);
    if (first) map[(size_t)b * T + w] = s;
}

// ---------------------------------------------------------------------------
// Kernel 3: one 256-thread block per output row (b, w).
//   mapped   -> DMA the 4 KB hidden row through LDS with the gfx1250 async
//               global<->LDS path (ASYNCcnt-tracked, no VGPR data movement).
//   unmapped -> non-temporal 128-bit zero fill (output is write-once).
// Row bases are uniform per block -> SGPR saddr; per-lane byte offset in VGPR
// (GVS addressing mode). Each lane owns a private 16 B LDS slot, so only the
// asynccnt wait is needed between the load and the store (no barrier).
// ---------------------------------------------------------------------------
__global__ void wp_write_rows(const float* __restrict__ hidden,
                              const int* __restrict__ map,
                              float* __restrict__ out,
                              int S, int T, int D) {
    __shared__ __align__(16) float lds_buf[256 * 4];
    const int row = blockIdx.x;            // in [0, B*T)
    const int tid = threadIdx.x;
    const int s   = map[row];              // uniform per block
    const int b   = row / T;

    float* dstp = out + (size_t)row * (size_t)D;

    if (s >= 0) {
        const float* srcp = hidden + ((size_t)b * (size_t)S + (size_t)s) * (size_t)D;
        // Low 32 bits of a generic pointer to LDS == LDS byte offset.
        uint32_t lds_off = (uint32_t)(uintptr_t)(&lds_buf[0]) + (uint32_t)tid * 16u;
        for (int base = tid * 4; base < D; base += blockDim.x * 4) {
            uint32_t goff = (uint32_t)base * 4u;  // byte offset within the row
            asm volatile(
                "global_load_async_to_lds_b128 %0, %1, %2\n\t"
                "s_wait_asynccnt 0\n\t"
                "global_store_async_from_lds_b128 %1, %0, %3\n\t"
                "s_wait_asynccnt 0"
                :
                : "v"(lds_off), "v"(goff), "s"(srcp), "s"(dstp)
                : "memory");
        }
    } else {
        const v4f z = {0.f, 0.f, 0.f, 0.f};
        for (int base = tid * 4; base < D; base += blockDim.x * 4) {
            __builtin_nontemporal_store(z, (v4f*)(dstp + base));
        }
    }
}

// ---------------------------------------------------------------------------
// Launch: inputs are [hidden_states f32 (B,S,D), word_offsets int (B,S),
// target_length int scalar]. Output: (B, T, D) f32. Workspace: B*T ints.
// ---------------------------------------------------------------------------
extern "C" void kernel_launch(void* const* d_in, const int* in_sizes, int n_in,
                              void* d_out, int out_size, void* d_ws, size_t ws_size,
                              hipStream_t stream) {
    const float* hidden = (const float*)d_in[0];
    const int*   ofs    = (const int*)d_in[1];
    float*       out    = (float*)d_out;
    int*         map    = (int*)d_ws;

    const int BS      = in_sizes[1];             // B*S
    const int D       = in_sizes[0] / in_sizes[1];
    const int B       = 8;                       // fixed by reference setup
    const int S       = BS / B;
    const int rowsOut = out_size / D;            // B*T
    const int T       = rowsOut / B;

    wp_init_map<<<(rowsOut + 255) / 256, 256, 0, stream>>>(map, rowsOut);
    wp_mark_first<<<(BS + 255) / 256, 256, 0, stream>>>(ofs, map, S, T, BS);
    wp_write_rows<<<rowsOut, 256, 0, stream>>>(hidden, map, out, S, T, D);
}